// HandmadeConv2d_82858509075082
// MI455X (gfx1250) — compile-verified
//
#include <hip/hip_runtime.h>

// ---- problem constants (from reference) ------------------------------------
#define N_IMG 32
#define C_IN  128
#define H_IN  64
#define W_IN  64
#define HW    (H_IN * W_IN)            // 4096
#define K_OUT 256
#define P_OUT 62
#define Q_OUT 62
#define PQ    (P_OUT * Q_OUT)          // 3844
#define KDIM  (C_IN * 9)               // 1152 = reduction dim (c,r,s)
#define NPIX  (N_IMG * PQ)             // 123008 = output pixels (n,p,q)

// ---- tiling ----------------------------------------------------------------
#define BM 64          // out-channels per workgroup
#define BN 128         // out-pixels per workgroup
#define KC 36          // K-chunk = lcm(4,9): (r,s) is compile-time per slot
#define NCHUNK (KDIM / KC)   // 32
#define APAIRS (BM * KC / 2)  // 1152 A K-pairs per chunk

typedef float v2f __attribute__((ext_vector_type(2)));
typedef float v8f __attribute__((ext_vector_type(8)));

__global__ __launch_bounds__(256)
void conv2d_implicit_gemm_f32wmma(const float* __restrict__ x,
                                  const float* __restrict__ w,
                                  float* __restrict__ out)
{
    // A: [m][k], rows padded to 38 floats (152 B) -> b64-aligned, conflict-free
    __shared__ __align__(16) float As[2][BM][KC + 2];
    // B: K-pair interleaved [k/2][n][2], odd pair-rows XOR-swizzled by 16 cols
    __shared__ __align__(16) float Bs[2][KC / 2][BN][2];

    const int tid  = threadIdx.x;        // 0..255 (8 waves, wave32)
    const int wid  = tid >> 5;
    const int lane = tid & 31;
    const int half = lane >> 4;          // 0: K pair (0,1) | 1: K pair (2,3)
    const int l16  = lane & 15;

    const int m_w = wid & 3;             // 4 x 16 = 64 M
    const int n_w = wid >> 2;            // 2 x 64 = 128 N

    const int m0 = blockIdx.y * BM;
    const int n0 = blockIdx.x * BN;

    // ---- A staging decomposition (loop-invariant): 1152 pairs, 256 threads --
    unsigned aRow[5], aKp[5], wpb[5];
    #pragma unroll
    for (int i = 0; i < 5; ++i) {
        const unsigned idx = (unsigned)(tid + 256 * i);
        const unsigned m   = idx / 18u;          // pair-row
        const unsigned kp  = idx - 18u * m;      // pair-col 0..17
        aRow[i] = m; aKp[i] = kp;
        wpb[i]  = (unsigned)(m0 + m) * KDIM + 2u * kp;
    }

    // ---- B staging decomposition: pixel nn, K-pairs p = 9g+i ----------------
    //      g is WAVE-UNIFORM; k mod 9 per slot is COMPILE-TIME (KC % 9 == 0).
    const int nn = tid & 127;
    const int g  = __builtin_amdgcn_readfirstlane(tid >> 7);   // 0 or 1
    const unsigned nidx = (unsigned)(n0 + nn);
    const unsigned nimg = nidx / PQ;
    const unsigned rem  = nidx - nimg * PQ;
    const unsigned p    = rem / Q_OUT;
    const unsigned q    = rem - p * Q_OUT;
    const unsigned xbase = nimg * (unsigned)(C_IN * HW) + p * W_IN + q
                         + (unsigned)(2 * g) * (unsigned)HW;   // fold c += 2g

    v2f areg[5];
    v2f breg[9];

    v8f acc[4] = { v8f{}, v8f{}, v8f{}, v8f{} };

    // ---- pipeline stages ----------------------------------------------------
    auto stage_load = [&](int kb) {           // global -> registers
        const int k0 = kb * KC;
        #pragma unroll
        for (int i = 0; i < 5; ++i)           // weights: contiguous b64 loads
            if (i < 4 || tid < 128)
                areg[i] = *(const v2f*)&w[wpb[i] + (unsigned)k0];
        // input gather: scalar chunk base + compile-time slot offset
        const unsigned cb = (unsigned)(4 * kb) * (unsigned)HW;  // c = 4*kb (+2g in xbase)
        #pragma unroll
        for (int i = 0; i < 9; ++i) {
            #pragma unroll
            for (int e = 0; e < 2; ++e) {
                const int t2 = 2 * i + e;                 // k mod 18, g-free
                const int rs = t2 % 9;                    // compile-time
                const int hi = t2 / 9;                    // compile-time carry
                const int co = hi * HW + (rs / 3) * W_IN + (rs % 3);
                breg[i][e] = x[xbase + cb + (unsigned)co];
            }
        }
    };
    auto stage_store = [&](int buf) {         // registers -> LDS (all b64)
        #pragma unroll
        for (int i = 0; i < 5; ++i)
            if (i < 4 || tid < 128)
                *(v2f*)&As[buf][aRow[i]][2 * aKp[i]] = areg[i];
        #pragma unroll
        for (int i = 0; i < 9; ++i) {
            const int pr = 9 * g + i;                     // pair-row
            const int cs = nn ^ ((pr & 1) << 4);          // bank swizzle
            *(v2f*)&Bs[buf][pr][cs][0] = breg[i];
        }
    };
    auto compute = [&](int buf) {             // 36 wmmas on the staged chunk
        const float* arow = &As[buf][m_w * 16 + l16][0];
        const int nb = n_w * 64;
        #pragma unroll
        for (int ks = 0; ks < KC; ks += 4) {
            const int koff = ks + half * 2;   // ISA A/B frag K striping
            const v2f a = *(const v2f*)(arow + koff);
            #pragma unroll
            for (int t = 0; t < 4; ++t) {
                const int cs = (nb + t * 16 + l16) ^ (half << 4);  // un-swizzle
                const v2f b = *(const v2f*)&Bs[buf][koff >> 1][cs][0];
                acc[t] = __builtin_amdgcn_wmma_f32_16x16x4_f32(
                             false, a, false, b, (short)0, acc[t], false, false);
            }
        }
    };

    // ---- software-pipelined main loop (double-buffered LDS) -----------------
    stage_load(0);
    stage_store(0);
    __syncthreads();

    for (int kb = 0; kb < NCHUNK; ++kb) {
        const int cur = kb & 1;
        if (kb + 1 < NCHUNK) stage_load(kb + 1);          // issue global loads
        compute(cur);                                     // wmma under latency
        if (kb + 1 < NCHUNK) stage_store(cur ^ 1);        // fill other buffer
        __syncthreads();
    }

    // ---- store D: VGPR j = row j (lanes 0-15) / row 8+j (lanes 16-31) -------
    const unsigned mbase = (unsigned)(m0 + m_w * 16 + half * 8);
    #pragma unroll
    for (int t = 0; t < 4; ++t) {
        const unsigned ni = (unsigned)(n0 + n_w * 64 + t * 16 + l16);
        const unsigned nI = ni / PQ;
        const unsigned rm = ni - nI * PQ;
        const unsigned ob = (nI * K_OUT + mbase) * PQ + rm;
        #pragma unroll
        for (int j = 0; j < 8; ++j)
            out[ob + (unsigned)j * PQ] = acc[t][j];
    }
}

extern "C" void kernel_launch(void* const* d_in, const int* in_sizes, int n_in,
                              void* d_out, int out_size, void* d_ws, size_t ws_size,
                              hipStream_t stream) {
    const float* x = (const float*)d_in[0];   // (32,128,64,64) f32
    const float* w = (const float*)d_in[1];   // (256,128,3,3)  f32
    float* out     = (float*)d_out;           // (32,256,62,62) f32

    dim3 grid(NPIX / BN, K_OUT / BM);         // (961, 4)
    dim3 block(256);
    conv2d_implicit_gemm_f32wmma<<<grid, block, 0, stream>>>(x, w, out);
}